// GNNGraphClassifier_29892972380773
// MI455X (gfx1250) — compile-verified
//
#include <hip/hip_runtime.h>
#include <math.h>

// ---------------------------------------------------------------------------
// GNN graph classifier forward (MI455X / gfx1250, wave32).
// Memory-bound scatter/gather workload; node-feature linear transforms use
// V_WMMA_F32_16X16X4_F32 (16-row tiles, two K=4 steps for K=8, Wl+Wr fused).
// ---------------------------------------------------------------------------

constexpr int   NNODES  = 200000;
constexpr int   NEDGES  = 3200000;
constexpr int   NGRAPHS = 1000;
constexpr int   ETOT    = NEDGES + NNODES;   // edges + self loops
constexpr float EPSL    = 1e-5f;
constexpr float NEG     = 0.2f;

typedef __attribute__((ext_vector_type(2))) float v2f;
typedef __attribute__((ext_vector_type(8))) float v8f;

#define DEVFN __device__ __forceinline__

DEVFN void atomicMaxF(float* addr, float val) {
  // works for all finite floats given init = -inf
  if (val >= 0.f) atomicMax((int*)addr, __float_as_int(val));
  else            atomicMin((unsigned int*)addr, __float_as_uint(val));
}

DEVFN void ln_tanh8(float* t, const float* __restrict__ g, const float* __restrict__ be) {
  float m = 0.f;
#pragma unroll
  for (int c = 0; c < 8; ++c) m += t[c];
  m *= 0.125f;
  float v = 0.f;
#pragma unroll
  for (int c = 0; c < 8; ++c) { float d = t[c] - m; v += d * d; }
  v *= 0.125f;
  float inv = rsqrtf(v + EPSL);
#pragma unroll
  for (int c = 0; c < 8; ++c) t[c] = tanhf((t[c] - m) * inv * g[c] + be[c]);
}

DEVFN void load_row8(const float* __restrict__ base, int row, float* t) {
  // rows are 32B and 16B-aligned in the workspace: two b128 loads
  const float4* p = (const float4*)(base) + row * 2;
  float4 a = p[0], b = p[1];
  t[0] = a.x; t[1] = a.y; t[2] = a.z; t[3] = a.w;
  t[4] = b.x; t[5] = b.y; t[6] = b.z; t[7] = b.w;
}

// ------------------------------- utility -----------------------------------
__global__ void k_fill(float* __restrict__ p, long long n, float v) {
  long long i = (long long)blockIdx.x * blockDim.x + threadIdx.x;
  long long st = (long long)gridDim.x * blockDim.x;
  for (; i < n; i += st) p[i] = v;
}

__global__ void k_reduce_sum(const float* __restrict__ v, long long n, float* __restrict__ out) {
  __shared__ float sm[256];
  long long i = (long long)blockIdx.x * blockDim.x + threadIdx.x;
  long long st = (long long)gridDim.x * blockDim.x;
  float acc = 0.f;
  for (; i < n; i += st) acc += v[i];
  sm[threadIdx.x] = acc;
  __syncthreads();
  for (int s = 128; s > 0; s >>= 1) {
    if ((int)threadIdx.x < s) sm[threadIdx.x] += sm[threadIdx.x + s];
    __syncthreads();
  }
  if (threadIdx.x == 0) atomicAdd(out, sm[0]);
}

// ------------------------- input transform (3->8) ---------------------------
__global__ void k_input(const float* __restrict__ x, const float* __restrict__ W,
                        const float* __restrict__ b, const float* __restrict__ g,
                        const float* __restrict__ be, float* __restrict__ h) {
  int n = blockIdx.x * blockDim.x + threadIdx.x;
  if (n >= NNODES) return;
  float t[8];
#pragma unroll
  for (int j = 0; j < 8; ++j) {
    float a = b[j];
#pragma unroll
    for (int i = 0; i < 3; ++i) a += x[n * 3 + i] * W[i * 8 + j];
    t[j] = a;
  }
  ln_tanh8(t, g, be);
#pragma unroll
  for (int j = 0; j < 8; ++j) h[n * 8 + j] = t[j];
}

// ------- fused node linears Yl = X@Wl + bl, Yr = X@Wr + br via WMMA ---------
// One wave per 16-row tile. K=8 split into two V_WMMA_F32_16X16X4_F32 steps.
// A 16x4 f32 layout: lanes 0-15 = rows; VGPR0 holds K={0,2} by lane-half,
// VGPR1 holds K={1,3}. B 4x16 mirrors the K split; cols 8..15 masked to zero
// branchlessly (loads always in-bounds -> no EXEC manipulation near WMMA).
__global__ void k_linear8_dual_wmma(const float* __restrict__ X,
                                    const float* __restrict__ Wl, const float* __restrict__ bl,
                                    const float* __restrict__ Wr, const float* __restrict__ br,
                                    float* __restrict__ Yl, float* __restrict__ Yr,
                                    int nrows) {
  int gw   = (blockIdx.x * blockDim.x + threadIdx.x) >> 5;  // global wave id
  int lane = threadIdx.x & 31;
  int row0 = gw * 16;
  if (row0 >= nrows) return;          // wave-uniform: EXEC all-1s inside
  int half = lane >> 4;               // 0: lanes 0-15, 1: lanes 16-31
  int m    = lane & 15;
  int m8   = m & 7;
  float msk = (m < 8) ? 1.f : 0.f;

  v2f a[2], wbl[2], wbr[2];
#pragma unroll
  for (int s = 0; s < 2; ++s) {
    int k0 = s * 4 + half * 2;
    a[s] = ((const v2f*)X)[(row0 + m) * 4 + (k0 >> 1)];  // one b64 load
    wbl[s].x = Wl[k0 * 8 + m8] * msk;
    wbl[s].y = Wl[(k0 + 1) * 8 + m8] * msk;
    wbr[s].x = Wr[k0 * 8 + m8] * msk;
    wbr[s].y = Wr[(k0 + 1) * 8 + m8] * msk;
  }
  v8f cl = {}, cr = {};
#pragma unroll
  for (int s = 0; s < 2; ++s) {       // interleave to relax D->C RAW chains
    cl = __builtin_amdgcn_wmma_f32_16x16x4_f32(false, a[s], false, wbl[s],
                                               (short)0, cl, false, false);
    cr = __builtin_amdgcn_wmma_f32_16x16x4_f32(false, a[s], false, wbr[s],
                                               (short)0, cr, false, false);
  }
  if (m < 8) {
    float bvl = bl[m8], bvr = br[m8];
#pragma unroll
    for (int r = 0; r < 8; ++r) {     // C: VGPR r -> row r + 8*half
      int row = row0 + r + half * 8;
      Yl[row * 8 + m8] = cl[r] + bvl;
      Yr[row * 8 + m8] = cr[r] + bvr;
    }
  }
}

// ----------------------------- GATv2 passes ---------------------------------
__global__ void k_gat_alpha(const float* __restrict__ xl, const float* __restrict__ xr,
                            const float* __restrict__ e, const int* __restrict__ src,
                            const int* __restrict__ dst, const float* __restrict__ We,
                            const float* __restrict__ att, const float* __restrict__ scal,
                            float* __restrict__ alpha, float* __restrict__ amax) {
  int i = blockIdx.x * blockDim.x + threadIdx.x;
  if (i >= ETOT) return;
  int s, d; float ev;
  if (i < NEDGES) { s = src[i]; d = dst[i]; ev = e[i]; }
  else            { s = d = i - NEDGES;     ev = scal[0] * (1.0f / NEDGES); }
  float lv[8], rv[8];
  load_row8(xl, s, lv);
  load_row8(xr, d, rv);
  float acc = 0.f;
#pragma unroll
  for (int c = 0; c < 8; ++c) {
    float mm = lv[c] + rv[c] + ev * We[c];
    mm = (mm > 0.f) ? mm : NEG * mm;       // leaky_relu
    acc += mm * att[c];
  }
  alpha[i] = acc;
  atomicMaxF(&amax[d], acc);
}

__global__ void k_gat_exp(float* __restrict__ alpha, const float* __restrict__ amax,
                          const int* __restrict__ dst, float* __restrict__ den) {
  int i = blockIdx.x * blockDim.x + threadIdx.x;
  if (i >= ETOT) return;
  int d = (i < NEDGES) ? dst[i] : (i - NEDGES);
  float ex = __expf(alpha[i] - amax[d]);
  alpha[i] = ex;
  atomicAdd(&den[d], ex);
}

__global__ void k_gat_aggr(const float* __restrict__ alpha, const float* __restrict__ den,
                           const float* __restrict__ xl, const int* __restrict__ src,
                           const int* __restrict__ dst, float* __restrict__ agg) {
  int i = blockIdx.x * blockDim.x + threadIdx.x;
  if (i >= ETOT) return;
  int s, d;
  if (i < NEDGES) { s = src[i]; d = dst[i]; }
  else            { s = d = i - NEDGES; }
  float w = alpha[i] / (den[d] + 1e-16f);
  float lv[8];
  load_row8(xl, s, lv);
#pragma unroll
  for (int c = 0; c < 8; ++c) atomicAdd(&agg[d * 8 + c], lv[c] * w);
}

// y = tanh(agg + bias); accumulate sum / sumsq for graph_norm
__global__ void k_tanh_stats(const float* __restrict__ agg, const float* __restrict__ bias,
                             float* __restrict__ y, float* __restrict__ sums) {
  __shared__ float s1[256], s2[256];
  long long i = (long long)blockIdx.x * blockDim.x + threadIdx.x;
  float v = 0.f;
  if (i < (long long)NNODES * 8) {
    v = tanhf(agg[i] + bias[i & 7]);
    y[i] = v;
  }
  s1[threadIdx.x] = v;
  s2[threadIdx.x] = v * v;
  __syncthreads();
  for (int s = 128; s > 0; s >>= 1) {
    if ((int)threadIdx.x < s) { s1[threadIdx.x] += s1[threadIdx.x + s];
                                s2[threadIdx.x] += s2[threadIdx.x + s]; }
    __syncthreads();
  }
  if (threadIdx.x == 0) { atomicAdd(&sums[0], s1[0]); atomicAdd(&sums[1], s2[0]); }
}

__global__ void k_gnorm_apply(const float* __restrict__ y, const float* __restrict__ g,
                              const float* __restrict__ be, const float* __restrict__ sums,
                              float* __restrict__ h) {
  long long i = (long long)blockIdx.x * blockDim.x + threadIdx.x;
  if (i >= (long long)NNODES * 8) return;
  const float invC = 1.0f / ((float)NNODES * 8.0f);
  float m   = sums[0] * invC;
  float var = sums[1] * invC - m * m;
  float inv = rsqrtf(var + EPSL);
  int c = (int)(i & 7);
  h[i] = (y[i] - m) * inv * g[c] + be[c];
}

__global__ void k_add_bias(const float* __restrict__ agg, const float* __restrict__ bias,
                           float* __restrict__ o) {
  long long i = (long long)blockIdx.x * blockDim.x + threadIdx.x;
  if (i >= (long long)NNODES * 8) return;
  o[i] = agg[i] + bias[i & 7];
}

__global__ void k_residual(const float* __restrict__ a, const float* __restrict__ b,
                           float* __restrict__ o) {
  long long i = (long long)blockIdx.x * blockDim.x + threadIdx.x;
  if (i >= (long long)NNODES * 8) return;
  o[i] = a[i] + b[i];
}

// --------------------------- per-edge MLP -----------------------------------
__global__ void k_edge_net(const float* __restrict__ h, const int* __restrict__ src,
                           const int* __restrict__ dst,
                           const float* __restrict__ W0, const float* __restrict__ b0,
                           const float* __restrict__ W1, const float* __restrict__ b1,
                           const float* __restrict__ W2, const float* __restrict__ b2,
                           const float* __restrict__ W3, const float* __restrict__ b3,
                           const float* __restrict__ g0, const float* __restrict__ be0,
                           const float* __restrict__ g1, const float* __restrict__ be1,
                           const float* __restrict__ g2, const float* __restrict__ be2,
                           float* __restrict__ e) {
  int i = blockIdx.x * blockDim.x + threadIdx.x;
  if (i >= NEDGES) return;
  int s = src[i], d = dst[i];
  float in[17];
  load_row8(h, s, in);
  load_row8(h, d, in + 8);
  in[16] = e[i];
  float t[8], u[8];
#pragma unroll
  for (int j = 0; j < 8; ++j) {
    float a = b0[j];
#pragma unroll
    for (int k = 0; k < 17; ++k) a += in[k] * W0[k * 8 + j];
    t[j] = a;
  }
  ln_tanh8(t, g0, be0);
#pragma unroll
  for (int j = 0; j < 8; ++j) {
    float a = b1[j];
#pragma unroll
    for (int k = 0; k < 8; ++k) a += t[k] * W1[k * 8 + j];
    u[j] = a;
  }
  ln_tanh8(u, g1, be1);
#pragma unroll
  for (int j = 0; j < 8; ++j) {
    float a = b2[j];
#pragma unroll
    for (int k = 0; k < 8; ++k) a += u[k] * W2[k * 8 + j];
    t[j] = a;
  }
  ln_tanh8(t, g2, be2);
  float o = b3[0];
#pragma unroll
  for (int k = 0; k < 8; ++k) o += t[k] * W3[k];
  e[i] = 1.f / (1.f + __expf(-o));
}

// ----------------------- pooling + prediction head --------------------------
__global__ void k_pool(const float* __restrict__ h, const int* __restrict__ batch,
                       float* __restrict__ sbuf) {
  int n = blockIdx.x * blockDim.x + threadIdx.x;
  if (n >= NNODES) return;
  int g = batch[n];
  float hv[8];
  load_row8(h, n, hv);
#pragma unroll
  for (int c = 0; c < 8; ++c) atomicAdd(&sbuf[g * 8 + c], hv[c]);
}

__global__ void k_pred(const float* __restrict__ sbuf,
                       const float* __restrict__ W0, const float* __restrict__ b0,
                       const float* __restrict__ W1, const float* __restrict__ b1,
                       const float* __restrict__ W2, const float* __restrict__ b2,
                       const float* __restrict__ W3, const float* __restrict__ b3,
                       const float* __restrict__ g0, const float* __restrict__ be0,
                       const float* __restrict__ g1, const float* __restrict__ be1,
                       const float* __restrict__ g2, const float* __restrict__ be2,
                       float* __restrict__ out) {
  int gi = blockIdx.x * blockDim.x + threadIdx.x;
  if (gi >= NGRAPHS) return;
  float t[8], u[8];
#pragma unroll
  for (int c = 0; c < 8; ++c) t[c] = sbuf[gi * 8 + c];
#pragma unroll
  for (int j = 0; j < 8; ++j) {
    float a = b0[j];
#pragma unroll
    for (int k = 0; k < 8; ++k) a += t[k] * W0[k * 8 + j];
    u[j] = a;
  }
  ln_tanh8(u, g0, be0);
#pragma unroll
  for (int j = 0; j < 8; ++j) {
    float a = b1[j];
#pragma unroll
    for (int k = 0; k < 8; ++k) a += u[k] * W1[k * 8 + j];
    t[j] = a;
  }
  ln_tanh8(t, g1, be1);
#pragma unroll
  for (int j = 0; j < 8; ++j) {
    float a = b2[j];
#pragma unroll
    for (int k = 0; k < 8; ++k) a += t[k] * W2[k * 8 + j];
    u[j] = a;
  }
  ln_tanh8(u, g2, be2);
  float o = b3[0];
#pragma unroll
  for (int k = 0; k < 8; ++k) o += u[k] * W3[k];
  out[gi] = o;
}

// ---------------------------------------------------------------------------
// Host launcher. Input pointer order assumes recursive dict-insertion-order
// flattening of setup_inputs():
//   0 x, 1 e, 2 edge_index, 3 batch,
//   4..7   inp.lin{W,b}, inp.ln{g,be}
//   8..15  edge.lins[0..3]{W,b}
//   16..21 edge.lns[0..2]{g,be}
//   22..49 node.gats[0..3]{Wl,bl,Wr,br,We,att,bias}
//   50..55 node.lns[0..2]{g,be}
//   56..63 pred.lins[0..3]{W,b}
//   64..69 pred.lns[0..2]{g,be}
// ---------------------------------------------------------------------------
extern "C" void kernel_launch(void* const* d_in, const int* in_sizes, int n_in,
                              void* d_out, int out_size, void* d_ws, size_t ws_size,
                              hipStream_t stream) {
  auto F = [&](int i) { return (const float*)d_in[i]; };

  const float* x     = F(0);
  const float* e_in  = F(1);
  const int*   ei    = (const int*)d_in[2];
  const int*   batch = (const int*)d_in[3];
  const int*   src   = ei;
  const int*   dst   = ei + NEDGES;

  int p = 4;
  const float* inp_W  = F(p++); const float* inp_b  = F(p++);
  const float* inp_g  = F(p++); const float* inp_be = F(p++);

  const float *eW[4], *eb[4];
  for (int k = 0; k < 4; ++k) { eW[k] = F(p++); eb[k] = F(p++); }
  const float *eg[3], *ebe[3];
  for (int k = 0; k < 3; ++k) { eg[k] = F(p++); ebe[k] = F(p++); }

  struct GatP { const float *Wl, *bl, *Wr, *br, *We, *att, *bias; } gat[4];
  for (int k = 0; k < 4; ++k) {
    gat[k].Wl = F(p++); gat[k].bl = F(p++); gat[k].Wr = F(p++); gat[k].br = F(p++);
    gat[k].We = F(p++); gat[k].att = F(p++); gat[k].bias = F(p++);
  }
  const float *ng[3], *nbe[3];
  for (int k = 0; k < 3; ++k) { ng[k] = F(p++); nbe[k] = F(p++); }

  const float *pW[4], *pb[4];
  for (int k = 0; k < 4; ++k) { pW[k] = F(p++); pb[k] = F(p++); }
  const float *pg[3], *pbe[3];
  for (int k = 0; k < 3; ++k) { pg[k] = F(p++); pbe[k] = F(p++); }

  float* out = (float*)d_out;

  // workspace layout (floats); all blocks are 32B-aligned (multiples of 8 floats)
  float* ws    = (float*)d_ws;
  float* h     = ws;
  float* h0    = h    + (size_t)NNODES * 8;
  float* hnn   = h0   + (size_t)NNODES * 8;
  float* xl    = hnn  + (size_t)NNODES * 8;
  float* xr    = xl   + (size_t)NNODES * 8;
  float* agg   = xr   + (size_t)NNODES * 8;
  float* ecur  = agg  + (size_t)NNODES * 8;
  float* alpha = ecur + (size_t)NEDGES;
  float* amax  = alpha + (size_t)ETOT;
  float* den   = amax + (size_t)NNODES;
  float* sbuf  = den  + (size_t)NNODES;
  float* scal  = sbuf + (size_t)NGRAPHS * 8;   // [0]=e_sum, [1]=gn_sum, [2]=gn_sumsq

  const int T = 256;
  auto NB = [](long long n) { return (int)((n + 255) / 256); };
  const int NB_WMMA = NB((long long)((NNODES + 15) / 16) * 32);

  (void)hipMemcpyAsync(ecur, e_in, sizeof(float) * NEDGES, hipMemcpyDeviceToDevice, stream);
  k_input<<<NB(NNODES), T, 0, stream>>>(x, inp_W, inp_b, inp_g, inp_be, h);

  for (int it = 0; it < 3; ++it) {
    (void)hipMemcpyAsync(h0, h, sizeof(float) * (size_t)NNODES * 8,
                         hipMemcpyDeviceToDevice, stream);
    // mean(e) for self-loop fill value
    k_fill<<<1, 32, 0, stream>>>(scal, 3, 0.f);
    k_reduce_sum<<<1024, T, 0, stream>>>(ecur, NEDGES, scal);

    for (int gi = 0; gi < 4; ++gi) {
      const GatP& gp = gat[gi];
      k_fill<<<512, T, 0, stream>>>(agg, (long long)NNODES * 8, 0.f);
      k_fill<<<256, T, 0, stream>>>(amax, NNODES, -INFINITY);
      k_fill<<<256, T, 0, stream>>>(den, NNODES, 0.f);
      k_linear8_dual_wmma<<<NB_WMMA, T, 0, stream>>>(h, gp.Wl, gp.bl, gp.Wr, gp.br,
                                                     xl, xr, NNODES);
      k_gat_alpha<<<NB(ETOT), T, 0, stream>>>(xl, xr, ecur, src, dst, gp.We, gp.att,
                                              scal, alpha, amax);
      k_gat_exp<<<NB(ETOT), T, 0, stream>>>(alpha, amax, dst, den);
      k_gat_aggr<<<NB(ETOT), T, 0, stream>>>(alpha, den, xl, src, dst, agg);
      if (gi < 3) {
        k_fill<<<1, 32, 0, stream>>>(scal + 1, 2, 0.f);
        k_tanh_stats<<<NB((long long)NNODES * 8), T, 0, stream>>>(agg, gp.bias, hnn, scal + 1);
        k_gnorm_apply<<<NB((long long)NNODES * 8), T, 0, stream>>>(hnn, ng[gi], nbe[gi],
                                                                   scal + 1, h);
      } else {
        k_add_bias<<<NB((long long)NNODES * 8), T, 0, stream>>>(agg, gp.bias, hnn);
      }
    }
    // edge MLP updates e in place (reads h_nn = hnn, old e)
    k_edge_net<<<NB(NEDGES), T, 0, stream>>>(hnn, src, dst,
                                             eW[0], eb[0], eW[1], eb[1], eW[2], eb[2],
                                             eW[3], eb[3],
                                             eg[0], ebe[0], eg[1], ebe[1], eg[2], ebe[2],
                                             ecur);
    // residual
    k_residual<<<NB((long long)NNODES * 8), T, 0, stream>>>(hnn, h0, h);
  }

  // pooling + prediction head
  k_fill<<<64, T, 0, stream>>>(sbuf, (long long)NGRAPHS * 8, 0.f);
  k_pool<<<NB(NNODES), T, 0, stream>>>(h, batch, sbuf);
  k_pred<<<NB(NGRAPHS), T, 0, stream>>>(sbuf,
                                        pW[0], pb[0], pW[1], pb[1], pW[2], pb[2],
                                        pW[3], pb[3],
                                        pg[0], pbe[0], pg[1], pbe[1], pg[2], pbe[2],
                                        out);
}